// MemoryBank_89489938580008
// MI455X (gfx1250) — compile-verified
//
#include <hip/hip_runtime.h>

// ---------------------------------------------------------------------------
// Bulk table copy (256 MiB) via CDNA5 async global<->LDS DMA path, with a
// double-buffered software pipeline:
//   prologue: 8x async loads -> drain
//   steady state per tile: 8x async stores (cur) + 8x async loads (next)
//                          -> single s_wait_asynccnt 0 drain
// Non-temporal hints keep the 512 MB stream from churning the 192 MB L2.
// ---------------------------------------------------------------------------
__device__ __forceinline__ void async_load8(unsigned lds,
                                            const float4* __restrict__ p) {
#pragma unroll
  for (int k = 0; k < 8; ++k)
    asm volatile("global_load_async_to_lds_b128 %0, %1, off th:TH_LOAD_NT"
                 :: "v"(lds + (unsigned)(k * 4096)), "v"(p + k * 256)
                 : "memory");
}

__device__ __forceinline__ void async_store8(float4* __restrict__ p,
                                             unsigned lds) {
#pragma unroll
  for (int k = 0; k < 8; ++k)
    asm volatile("global_store_async_from_lds_b128 %0, %1, off th:TH_STORE_NT"
                 :: "v"(p + k * 256), "v"(lds + (unsigned)(k * 4096))
                 : "memory");
}

__global__ __launch_bounds__(256) void mb_copy_async(
    const float4* __restrict__ src, float4* __restrict__ dst, long long n16) {
  __shared__ float4 smem[4096];  // 64 KB: two 32 KB staging buffers
  const int tid = threadIdx.x;
  // low 32 bits of a flat LDS address == byte offset within the WG allocation
  const unsigned ldsA = (unsigned)(size_t)(&smem[tid]);
  const unsigned ldsB = ldsA + 32768u;

  const long long TILE = 2048;  // float4 per block-tile (32 KB)
  const long long ntiles = n16 / TILE;
  const long long tstride = gridDim.x;

  long long t = blockIdx.x;
  if (t < ntiles) {
    async_load8(ldsA, src + t * TILE + tid);
    asm volatile("s_wait_asynccnt 0" ::: "memory");
    unsigned useA = 1;
    for (;;) {
      const unsigned cur = useA ? ldsA : ldsB;
      const unsigned nxt = useA ? ldsB : ldsA;
      async_store8(dst + t * TILE + tid, cur);
      const long long tn = t + tstride;
      if (tn < ntiles) async_load8(nxt, src + tn * TILE + tid);
      asm volatile("s_wait_asynccnt 0" ::: "memory");
      if (tn >= ntiles) break;
      t = tn;
      useA ^= 1;
    }
  }

  // Tail (n16 not multiple of TILE): plain b128 copy
  const long long tail = ntiles * TILE;
  const long long stride = (long long)gridDim.x * blockDim.x;
  for (long long i = tail + (long long)blockIdx.x * blockDim.x + tid; i < n16;
       i += stride)
    dst[i] = src[i];
}

// ---------------------------------------------------------------------------
// Sequential-duplicate-safe momentum update.
// One wave32 per batch element. Block i proceeds only if it is the FIRST
// occurrence of targets[i]; it then applies all matching updates j>=i in
// order (matches the lax.scan semantics). Lane l owns floats [8l, 8l+8) of
// the 256-wide row; the L2 norm is a wave32 shuffle reduction.
// ---------------------------------------------------------------------------
__global__ __launch_bounds__(32) void mb_update(
    const float* __restrict__ x, const int* __restrict__ tgt,
    float* __restrict__ out, int batch) {
  const int i = blockIdx.x;
  const int lane = threadIdx.x;
  const int y = tgt[i];

  // first-occurrence ownership check (lane-parallel scan of tgt[0..i))
  for (int j0 = 0; j0 < i; j0 += 32) {
    const int jj = j0 + lane;
    const bool hit = (jj < i) && (tgt[jj] == y);
    if (__any(hit)) return;  // uniform across the wave
  }

  float* row = out + (size_t)y * 256 + lane * 8;
  float4 a = *(const float4*)(row);
  float4 b = *(const float4*)(row + 4);

  const float m = 0.1f, om = 0.9f, eps = 1e-12f;

  for (int j = i; j < batch; ++j) {
    if (tgt[j] != y) continue;  // uniform branch (scalar load of tgt[j])
    const float* xr = x + (size_t)j * 256 + lane * 8;
    const float4 xa = *(const float4*)(xr);
    const float4 xb = *(const float4*)(xr + 4);

    a.x = m * a.x + om * xa.x;  a.y = m * a.y + om * xa.y;
    a.z = m * a.z + om * xa.z;  a.w = m * a.w + om * xa.w;
    b.x = m * b.x + om * xb.x;  b.y = m * b.y + om * xb.y;
    b.z = m * b.z + om * xb.z;  b.w = m * b.w + om * xb.w;

    float ss = a.x * a.x + a.y * a.y + a.z * a.z + a.w * a.w +
               b.x * b.x + b.y * b.y + b.z * b.z + b.w * b.w;
    ss += __shfl_xor(ss, 16, 32);
    ss += __shfl_xor(ss, 8, 32);
    ss += __shfl_xor(ss, 4, 32);
    ss += __shfl_xor(ss, 2, 32);
    ss += __shfl_xor(ss, 1, 32);

    const float inv = 1.0f / fmaxf(sqrtf(ss), eps);
    a.x *= inv; a.y *= inv; a.z *= inv; a.w *= inv;
    b.x *= inv; b.y *= inv; b.z *= inv; b.w *= inv;
  }

  *(float4*)(row) = a;
  *(float4*)(row + 4) = b;
}

extern "C" void kernel_launch(void* const* d_in, const int* in_sizes, int n_in,
                              void* d_out, int out_size, void* d_ws,
                              size_t ws_size, hipStream_t stream) {
  const float* inputs = (const float*)d_in[0];    // [1024, 256] f32
  const int* targets = (const int*)d_in[1];       // [1024] int
  const float* feats = (const float*)d_in[2];     // [262144, 256] f32
  float* out = (float*)d_out;                     // [262144, 256] f32

  const long long n16 = (long long)in_sizes[2] / 4;  // float4 count
  const long long ntiles = n16 / 2048;
  long long want = ntiles > 0 ? ntiles : 1;
  int cblocks = (int)(want < 2048 ? want : 2048);

  mb_copy_async<<<cblocks, 256, 0, stream>>>((const float4*)feats,
                                             (float4*)out, n16);

  const int batch = in_sizes[1];
  mb_update<<<batch, 32, 0, stream>>>(inputs, targets, out, batch);
}